// MoE_46617575031150
// MI455X (gfx1250) — compile-verified
//
#include <hip/hip_runtime.h>
#include <math.h>

// ---------------------------------------------------------------------------
// MoE (4 routed experts top-2 + 2 shared experts), fp32, for MI455X (gfx1250)
//
//   1) gate_topk: per-token gating -> dense rscale[T][E] (softmax weight for
//      top-2 experts, 0 otherwise)
//   2) per expert pass (6 total):  moe_gemm mode0:  H = relu(X*W1 + b1)
//                                  moe_gemm mode1:  OUT (+)= s * (H*W2 + b2)
//   GEMMs use V_WMMA_F32_16X16X4_F32 (fp32 matrix pipe, exact fp32 accumulate
//   -> matches fp32 reference numerics).
//
//   This revision software-pipelines both levels of the memory hierarchy:
//     - global->LDS: next BK tile is loaded into registers before the compute
//       phase and stored to LDS after the post-compute barrier, so the
//       s_wait_loadcnt lands behind 64 WMMAs of work.
//     - LDS->VGPR: fragments are double-buffered; k-step s+4's ds_loads are
//       issued before k-step s's WMMAs, removing the ds_load -> s_wait_dscnt
//       -> wmma serialization seen in the previous build.
// ---------------------------------------------------------------------------

typedef __attribute__((ext_vector_type(2))) float v2f;
typedef __attribute__((ext_vector_type(8))) float v8f;

#define TOK_TOTAL 8192   // B*S
#define DDIM 1024
#define FDIM 4096
#define ODIM 1024
#define NEXP 4
#define NSHARED 2

#define BM 128
#define BN 128
#define BK 32
#define PAD 4            // rows stay 16B-aligned (36/132 floats) + conflict-free

// ---------------------------------------------------------------------------
// Gating: one wave per token. logits = x[t]@gate_w + gate_b ; softmax over 4;
// top-2 (ties -> lowest index, matching jax.lax.top_k); write dense scales.
// ---------------------------------------------------------------------------
__global__ __launch_bounds__(256)
void gate_topk(const float* __restrict__ x, const float* __restrict__ gw,
               const float* __restrict__ gb, float* __restrict__ rscale)
{
    const int lane = threadIdx.x & 31;
    const int wave = threadIdx.x >> 5;
    const int t = blockIdx.x * 8 + wave;

    const float* xr = x + (size_t)t * DDIM;
    float a0 = 0.f, a1 = 0.f, a2 = 0.f, a3 = 0.f;
    for (int d = lane; d < DDIM; d += 32) {
        const float  xv = xr[d];
        const float4 g  = *(const float4*)(gw + (size_t)d * NEXP);
        a0 += xv * g.x; a1 += xv * g.y; a2 += xv * g.z; a3 += xv * g.w;
    }
    for (int off = 16; off; off >>= 1) {
        a0 += __shfl_xor(a0, off, 32);
        a1 += __shfl_xor(a1, off, 32);
        a2 += __shfl_xor(a2, off, 32);
        a3 += __shfl_xor(a3, off, 32);
    }
    if (lane == 0) {
        float lg[NEXP] = { a0 + gb[0], a1 + gb[1], a2 + gb[2], a3 + gb[3] };
        float m = lg[0];
        for (int e = 1; e < NEXP; ++e) m = lg[e] > m ? lg[e] : m;
        float w[NEXP], s = 0.f;
        for (int e = 0; e < NEXP; ++e) { w[e] = expf(lg[e] - m); s += w[e]; }
        const float inv = 1.f / s;
        for (int e = 0; e < NEXP; ++e) w[e] *= inv;
        int i0 = 0;
        for (int e = 1; e < NEXP; ++e) if (w[e] > w[i0]) i0 = e;
        int i1 = -1;
        for (int e = 0; e < NEXP; ++e)
            if (e != i0 && (i1 < 0 || w[e] > w[i1])) i1 = e;
        float* rs = rscale + (size_t)t * NEXP;
        for (int e = 0; e < NEXP; ++e)
            rs[e] = (e == i0 || e == i1) ? w[e] : 0.f;
    }
}

// Fragment gather per the 16x16x4 f32 WMMA VGPR layouts:
//   A 16x4: lanes 0-15 hold K pair {ks,ks+1}, lanes 16-31 hold {ks+2,ks+3}
//   B 4x16: mirrored over the N columns
#define LOAD_FRAGS(ksrc, AF, BF)                                           \
    {                                                                      \
        _Pragma("unroll")                                                  \
        for (int i_ = 0; i_ < 2; ++i_) {                                   \
            const int row_ = wm + i_ * 16 + l16;                           \
            AF[i_].x = lsA[row_][(ksrc) + 2 * khalf + 0];                  \
            AF[i_].y = lsA[row_][(ksrc) + 2 * khalf + 1];                  \
        }                                                                  \
        _Pragma("unroll")                                                  \
        for (int j_ = 0; j_ < 4; ++j_) {                                   \
            const int col_ = wn + j_ * 16 + l16;                           \
            BF[j_].x = lsB[(ksrc) + 2 * khalf + 0][col_];                  \
            BF[j_].y = lsB[(ksrc) + 2 * khalf + 1][col_];                  \
        }                                                                  \
    }

// ---------------------------------------------------------------------------
// Blocked fp32 WMMA GEMM.  C = A[MxK] * W[KxN] (row-major, lda=K, ldb=ldc=N).
// mode 0: C = relu(acc + bias[col])                       (hidden layer)
// mode 1: C (+)= scale_row * (acc + bias[col])            (output layer)
//   scale_row = rowscale ? rowscale[row*rs_stride] : cscale
// Grid: (N/BN, M/BM); block: 256 threads = 8 waves (4 M x 2 N, 32x64 / wave).
// ---------------------------------------------------------------------------
__global__ __launch_bounds__(256)
void moe_gemm(const float* __restrict__ A, const float* __restrict__ W,
              const float* __restrict__ bias, float* __restrict__ C,
              const float* __restrict__ rowscale, int rs_stride, float cscale,
              int K, int N, int mode, int accumulate)
{
    __shared__ float lsA[BM][BK + PAD];   // 128 x 36 floats
    __shared__ float lsB[BK][BN + PAD];   // 32 x 132 floats

    const int tid   = threadIdx.x;
    const int lane  = tid & 31;
    const int wave  = tid >> 5;
    const int wm    = (wave & 3) * 32;    // wave M offset inside block
    const int wn    = (wave >> 2) * 64;   // wave N offset inside block
    const int khalf = lane >> 4;          // 0: K pair {0,1}, 1: K pair {2,3}
    const int l16   = lane & 15;

    const int bm0 = blockIdx.y * BM;
    const int bn0 = blockIdx.x * BN;

    v8f acc[2][4];
#pragma unroll
    for (int i = 0; i < 2; ++i)
#pragma unroll
        for (int j = 0; j < 4; ++j) acc[i][j] = (v8f){};

    const float* Ab = A + (size_t)bm0 * K;
    const float* Wb = W + bn0;

    // Register staging for the global->LDS pipeline.
    float4 ra[4], rb[4];
    auto load_tile = [&](int kk) {
#pragma unroll
        for (int t = 0; t < 4; ++t) {
            const int idx = tid + t * 256;
            const int r   = idx >> 3;          // 0..127
            const int c4  = (idx & 7) << 2;    // 0,4,...,28
            ra[t] = *(const float4*)(Ab + (size_t)r * K + kk + c4);
        }
#pragma unroll
        for (int t = 0; t < 4; ++t) {
            const int idx = tid + t * 256;
            const int r   = idx >> 5;          // 0..31
            const int c4  = (idx & 31) << 2;   // 0,4,...,124
            rb[t] = *(const float4*)(Wb + (size_t)(kk + r) * N + c4);
        }
    };
    auto store_tile = [&]() {
#pragma unroll
        for (int t = 0; t < 4; ++t) {
            const int idx = tid + t * 256;
            *(float4*)&lsA[idx >> 3][(idx & 7) << 2] = ra[t];
        }
#pragma unroll
        for (int t = 0; t < 4; ++t) {
            const int idx = tid + t * 256;
            *(float4*)&lsB[idx >> 5][(idx & 31) << 2] = rb[t];
        }
    };

    // Prologue: first tile into LDS.
    load_tile(0);
    store_tile();
    __syncthreads();

    for (int kk = 0; kk < K; kk += BK) {
        const bool last = (kk + BK >= K);
        if (!last) load_tile(kk + BK);   // global loads in flight during compute

        // Fragment-level software pipeline (double-buffered).
        v2f af[2][2], bf[2][4];
        LOAD_FRAGS(0, af[0], bf[0]);
#pragma unroll
        for (int ks = 0; ks < BK; ks += 4) {
            const int cur = (ks >> 2) & 1;
            const int nxt = cur ^ 1;
            if (ks + 4 < BK) LOAD_FRAGS(ks + 4, af[nxt], bf[nxt]);
#pragma unroll
            for (int i = 0; i < 2; ++i)
#pragma unroll
                for (int j = 0; j < 4; ++j)
                    acc[i][j] = __builtin_amdgcn_wmma_f32_16x16x4_f32(
                        false, af[cur][i], false, bf[cur][j], (short)0,
                        acc[i][j], false, false);
        }
        __syncthreads();
        if (!last) store_tile();         // waits for globals here, not earlier
        __syncthreads();
    }

    // Epilogue. C/D layout: VGPR r -> M = r + 8*khalf, N = l16.
#pragma unroll
    for (int i = 0; i < 2; ++i) {
#pragma unroll
        for (int j = 0; j < 4; ++j) {
            const int col = bn0 + wn + j * 16 + l16;
            const float bv = bias[col];
#pragma unroll
            for (int r = 0; r < 8; ++r) {
                const int row = bm0 + wm + i * 16 + khalf * 8 + r;
                const float v = acc[i][j][r] + bv;
                float* cp = C + (size_t)row * N + col;
                if (mode == 0) {
                    *cp = v > 0.f ? v : 0.f;
                } else {
                    const float s = rowscale ? rowscale[(size_t)row * rs_stride]
                                             : cscale;
                    float o = s * v;
                    if (accumulate) o += *cp;
                    *cp = o;
                }
            }
        }
    }
}

// ---------------------------------------------------------------------------
// Launcher
// ---------------------------------------------------------------------------
extern "C" void kernel_launch(void* const* d_in, const int* in_sizes, int n_in,
                              void* d_out, int out_size, void* d_ws, size_t ws_size,
                              hipStream_t stream)
{
    const float* x   = (const float*)d_in[0];   // [T, D]
    const float* gw  = (const float*)d_in[1];   // [D, E]
    const float* gb  = (const float*)d_in[2];   // [E]
    const float* sw1 = (const float*)d_in[3];   // [NS, D, F]
    const float* sb1 = (const float*)d_in[4];   // [NS, F]
    const float* sw2 = (const float*)d_in[5];   // [NS, F, O]
    const float* sb2 = (const float*)d_in[6];   // [NS, O]
    const float* rw1 = (const float*)d_in[7];   // [E, D, F]
    const float* rb1 = (const float*)d_in[8];   // [E, F]
    const float* rw2 = (const float*)d_in[9];   // [E, F, O]
    const float* rb2 = (const float*)d_in[10];  // [E, O]
    float* out = (float*)d_out;                 // [T, O]

    const int T = TOK_TOTAL;
    float* rscale = (float*)d_ws;               // T*E floats
    float* h      = rscale + (size_t)T * NEXP;  // tch*F floats

    // Pick largest token chunk whose hidden buffer fits in the workspace.
    const size_t head = (size_t)T * NEXP * sizeof(float);
    const size_t avail = ws_size > head ? ws_size - head : 0;
    int tch = 2048;
    while (tch > BM && (size_t)tch * FDIM * sizeof(float) > avail) tch >>= 1;

    gate_topk<<<T / 8, 256, 0, stream>>>(x, gw, gb, rscale);

    const dim3 blk(256);
    for (int c = 0; c < T; c += tch) {
        const dim3 g1(FDIM / BN, tch / BM);
        const dim3 g2(ODIM / BN, tch / BM);
        // Shared experts: average of NS MLPs -> scale 1/NS; first pass writes.
        for (int n = 0; n < NSHARED; ++n) {
            moe_gemm<<<g1, blk, 0, stream>>>(
                x + (size_t)c * DDIM, sw1 + (size_t)n * DDIM * FDIM,
                sb1 + (size_t)n * FDIM, h, nullptr, 0, 0.f, DDIM, FDIM, 0, 0);
            moe_gemm<<<g2, blk, 0, stream>>>(
                h, sw2 + (size_t)n * FDIM * ODIM, sb2 + (size_t)n * ODIM,
                out + (size_t)c * ODIM, nullptr, 0, 1.0f / NSHARED,
                FDIM, ODIM, 1, n > 0);
        }
        // Routed experts: dense compute, per-token scale (0 if not in top-2).
        for (int e = 0; e < NEXP; ++e) {
            moe_gemm<<<g1, blk, 0, stream>>>(
                x + (size_t)c * DDIM, rw1 + (size_t)e * DDIM * FDIM,
                rb1 + (size_t)e * FDIM, h, nullptr, 0, 0.f, DDIM, FDIM, 0, 0);
            moe_gemm<<<g2, blk, 0, stream>>>(
                h, rw2 + (size_t)e * FDIM * ODIM, rb2 + (size_t)e * ODIM,
                out + (size_t)c * ODIM, rscale + (size_t)c * NEXP + e, NEXP,
                0.f, FDIM, ODIM, 1, 1);
        }
    }
}